// DeQuantizer_63900523430425
// MI455X (gfx1250) — compile-verified
//
#include <hip/hip_runtime.h>
#include <stdint.h>

// ---------------------------------------------------------------------------
// GPTQ 4-bit dequantization for MI455X (gfx1250).
//   out[k][n] = scales[g_idx[k]][n] * (w4[k][n] - z4[g_idx[k]][n])
// Store-bandwidth bound: ~203 MB traffic -> ~8.7 us floor at 23.3 TB/s.
// gfx1250 specifics used:
//   * 3x TENSOR_LOAD_TO_LDS (Tensor Data Mover) + s_wait_tensorcnt for the
//     per-block scales / qzeros / g_idx tiles.
//   * wave32 sizing, LDS pair-table to keep the hot loop near the issue roof.
// Hot loop exploits sorted g_idx: the 8 rows of one packed word nearly always
// share a group -> one ds_load_b64 of {scale, zero} per 8 outputs.
// ---------------------------------------------------------------------------

#define KDIM    4096
#define NDIM    11008
#define GROUPS  32
#define PACK    8
#define KW      (KDIM / PACK)        // 512 packed rows
#define COLS    256                  // columns per block  (NDIM = 43 * 256)
#define KW_BLK  32                   // packed rows per block (KW = 16 * 32)
#define GRID_X  (NDIM / COLS)        // 43
#define GRID_Y  (KW / KW_BLK)        // 16
#define ZW_COLS (COLS / PACK)        // 32 qzeros words per group per tile

typedef unsigned int u32x4 __attribute__((ext_vector_type(4)));
typedef int          i32x4 __attribute__((ext_vector_type(4)));
typedef int          i32x8 __attribute__((ext_vector_type(8)));

#if __has_builtin(__builtin_amdgcn_tensor_load_to_lds)
#define HAVE_TDM 1
#endif

#ifdef HAVE_TDM
// Issue one TDM 2D tile load: tile0 x tile1 elements of 4-byte data,
// row stride `stride0` elements, from `gptr` into LDS byte offset `lds_off`.
// Descriptor layout per CDNA5 ISA sec. 8.3 (group 0) / 8.4 (group 1).
__device__ __forceinline__ void tdm_load_2d(const void* gptr, unsigned lds_off,
                                            unsigned dim0, unsigned dim1,
                                            unsigned tile0, unsigned tile1,
                                            unsigned stride0) {
  uint64_t ga = (uint64_t)gptr;
  u32x4 g0;
  g0.x = 1u;                                         // count=1, user descriptor
  g0.y = lds_off;                                    // lds_addr (bytes)
  g0.z = (unsigned)(ga & 0xffffffffu);               // global_addr[31:0]
  g0.w = (unsigned)((ga >> 32) & 0x1ffffffu)         // global_addr[56:32]
       | (2u << 30);                                 // type = 2 ("image")
  i32x8 g1;
  g1[0] = (int)(2u << 16);                           // data_size=2 (4 bytes)
  g1[1] = (int)((dim0 & 0xffffu) << 16);             // tensor_dim0[15:0]
  g1[2] = (int)(((dim0 >> 16) & 0xffffu) | ((dim1 & 0xffffu) << 16));
  g1[3] = (int)(((dim1 >> 16) & 0xffffu) | ((tile0 & 0xffffu) << 16));
  g1[4] = (int)(tile1 & 0xffffu);                    // tile_dim1; tile_dim2=0
  g1[5] = (int)stride0;                              // tensor_dim0_stride[31:0]
  g1[6] = 0;                                         // stride0[47:32], stride1 lo
  g1[7] = 0;                                         // stride1 hi (unused, 2D)
  i32x4 z4 = {0, 0, 0, 0};
#if __clang_major__ >= 23
  i32x8 z8 = {0, 0, 0, 0, 0, 0, 0, 0};
  __builtin_amdgcn_tensor_load_to_lds(g0, g1, z4, z4, z8, 0);
#else
  __builtin_amdgcn_tensor_load_to_lds(g0, g1, z4, z4, 0);
#endif
}
#endif

__global__ __launch_bounds__(256) void dequant4_kernel(
    const int*   __restrict__ qweight,   // [KW, NDIM]
    const int*   __restrict__ qzeros,    // [GROUPS, NDIM/8]
    const float* __restrict__ scales,    // [GROUPS, NDIM]
    const int*   __restrict__ g_idx,     // [KDIM]
    float*       __restrict__ out)       // [KDIM, NDIM]
{
  __shared__ float2 pair_tab[GROUPS * COLS];   // {scale, (float)zero} (64 KB)
  __shared__ float  s_stage[GROUPS * COLS];    // raw scales tile      (32 KB)
  __shared__ int    z_stage[GROUPS * ZW_COLS]; // raw qzeros words     ( 4 KB)
  __shared__ int    gv[KW_BLK * PACK];         // g_idx slice          ( 1 KB)

  const int t   = (int)threadIdx.x;
  const int n0  = (int)blockIdx.x * COLS;
  const int n   = n0 + t;
  const int kw0 = (int)blockIdx.y * KW_BLK;
  const int k0  = kw0 * PACK;

#ifdef HAVE_TDM
  if (t < 32) {
    // Wave 0 drives the Tensor Data Mover; other waves branch around.
    tdm_load_2d(&scales[n0], (unsigned)(size_t)(void*)s_stage,
                NDIM, GROUPS, COLS, GROUPS, NDIM);
    tdm_load_2d(&qzeros[n0 / PACK], (unsigned)(size_t)(void*)z_stage,
                NDIM / PACK, GROUPS, ZW_COLS, GROUPS, NDIM / PACK);
    tdm_load_2d(&g_idx[k0], (unsigned)(size_t)(void*)gv,
                KDIM, 1, KW_BLK * PACK, 1, KDIM);
    __builtin_amdgcn_s_wait_tensorcnt(0);
  }
#else
  gv[t] = g_idx[k0 + t];
  #pragma unroll 4
  for (int g = 0; g < GROUPS; ++g)
    s_stage[g * COLS + t] = scales[g * NDIM + n];
  for (int i = t; i < GROUPS * ZW_COLS; i += 256)
    z_stage[i] = qzeros[(i / ZW_COLS) * (NDIM / PACK) + n0 / PACK + (i % ZW_COLS)];
#endif
  __syncthreads();

  // Build the {scale, zero} pair table (float2 -> single ds_load_b64 later).
  {
    const int zw_idx = t >> 3;
    const int zshift = (t & 7) * 4;
    #pragma unroll 4
    for (int g = 0; g < GROUPS; ++g) {
      const float s  = s_stage[g * COLS + t];
      const int   zw = z_stage[g * ZW_COLS + zw_idx];
      const float zf = (float)((zw >> zshift) & 15);
      pair_tab[g * COLS + t] = make_float2(s, zf);
    }
  }
  __syncthreads();

  const int*  qrow = qweight + (size_t)kw0 * NDIM + n;
  float*      orow = out + (size_t)k0 * NDIM + n;

  for (int i = 0; i < KW_BLK; ++i) {
    const int w32 = __builtin_nontemporal_load(qrow + (size_t)i * NDIM);
    const int rb  = i * PACK;
    float* o = orow + (size_t)rb * NDIM;
    const int gfirst = gv[rb];
    const int glast  = gv[rb + PACK - 1];
    if (gfirst == glast) {
      // Sorted g_idx: all 8 rows in one group (common case, ~97% of words).
      const float2 p = pair_tab[gfirst * COLS + t];
      #pragma unroll
      for (int j = 0; j < PACK; ++j) {
        const float w = (float)((w32 >> (4 * j)) & 15);
        __builtin_nontemporal_store(p.x * (w - p.y), o + (size_t)j * NDIM);
      }
    } else {
      // Group boundary inside this packed word: per-row pair lookup.
      #pragma unroll
      for (int j = 0; j < PACK; ++j) {
        const int    g = gv[rb + j];
        const float2 p = pair_tab[g * COLS + t];
        const float  w = (float)((w32 >> (4 * j)) & 15);
        __builtin_nontemporal_store(p.x * (w - p.y), o + (size_t)j * NDIM);
      }
    }
  }
}

extern "C" void kernel_launch(void* const* d_in, const int* in_sizes, int n_in,
                              void* d_out, int out_size, void* d_ws, size_t ws_size,
                              hipStream_t stream) {
  (void)in_sizes; (void)n_in; (void)out_size; (void)d_ws; (void)ws_size;
  const int*   qweight = (const int*)d_in[0];   // [512, 11008] int32
  const int*   qzeros  = (const int*)d_in[1];   // [32, 1376]   int32
  const float* scales  = (const float*)d_in[2]; // [32, 11008]  fp32
  const int*   g_idx   = (const int*)d_in[3];   // [4096]       int32
  float*       out     = (float*)d_out;         // [4096, 11008] fp32

  dim3 grid(GRID_X, GRID_Y);
  dim3 block(256);
  dequant4_kernel<<<grid, block, 0, stream>>>(qweight, qzeros, scales, g_idx, out);
}